// Quadratic_77936476553490
// MI455X (gfx1250) — compile-verified
//
#include <hip/hip_runtime.h>

typedef float v2f __attribute__((ext_vector_type(2)));
typedef float v8f __attribute__((ext_vector_type(8)));

__device__ __forceinline__ float bperm(int idx, float v) {
    return __int_as_float(__builtin_amdgcn_ds_bpermute(idx, __float_as_int(v)));
}

// Merge two row-partial registers via raw ds_bpermute with precomputed byte
// index ((lane^k)<<2): lanes with (lane&k)==0 carry a's pair-sum, lanes with
// the bit set carry b's. 8 cross-lane ops for 8 full 16-lane row sums.
__device__ __forceinline__ float comb(float a, float b, bool hi, int idx) {
    float keep = hi ? b : a;
    float send = hi ? a : b;
    return keep + bperm(idx, send);
}

// WMMA (C initialized with broadcast W) + elementwise row-dot + reduce.
// Returns per-lane finished value: lane l -> row (l&7) + 8*(l>=16) of the tile.
__device__ __forceinline__ float tile_body(v2f a0, v2f a1, v2f a2,
                                           const float xc[8],
                                           v2f bq0, v2f bq1, v2f bq2,
                                           float Wl,
                                           bool h1, bool h2, bool h4,
                                           int i1, int i2, int i4, int i8) {
    v8f acc = {Wl, Wl, Wl, Wl, Wl, Wl, Wl, Wl};   // C[m][e] = W[e]
    acc = __builtin_amdgcn_wmma_f32_16x16x4_f32(false, a0, false, bq0,
                                                (short)0, acc, false, false);
    acc = __builtin_amdgcn_wmma_f32_16x16x4_f32(false, a1, false, bq1,
                                                (short)0, acc, false, false);
    acc = __builtin_amdgcn_wmma_f32_16x16x4_f32(false, a2, false, bq2,
                                                (short)0, acc, false, false);

    float p[8];
    #pragma unroll
    for (int r = 0; r < 8; ++r)
        p[r] = acc[r] * xc[r];                     // (T+W) ⊙ X

    float c0 = comb(p[0], p[1], h1, i1);
    float c1 = comb(p[2], p[3], h1, i1);
    float c2 = comb(p[4], p[5], h1, i1);
    float c3 = comb(p[6], p[7], h1, i1);
    float d0 = comb(c0, c1, h2, i2);
    float d1 = comb(c2, c3, h2, i2);
    float e  = comb(d0, d1, h4, i4);
    return e + bperm(i8, e);
}

__global__ void Quadratic_77936476553490_kernel(const float* __restrict__ X,
                                                const float* __restrict__ bsc,
                                                const float* __restrict__ W,
                                                const float* __restrict__ Q,
                                                float* __restrict__ out,
                                                int Nrows, int numTiles) {
    constexpr int D = 11;
    const int lane = threadIdx.x & 31;
    const int half = lane >> 4;
    const int l15  = lane & 15;
    const int wid  = (blockIdx.x * blockDim.x + threadIdx.x) >> 5;
    const int nW   = (gridDim.x * blockDim.x) >> 5;

    const bool colOK = (l15 < D);
    const int  colB  = colOK ? l15 : 0;

    // ---- B fragments: Q zero-padded to 12x16 (rows K=4k+2*half+{0,1}, N=l15) ----
    v2f bq0, bq1, bq2;
    {
        const int r0 = 0 + 2 * half;
        const int r1 = 4 + 2 * half;
        const int r2 = 8 + 2 * half;                   // 8 or 10
        bq0.x = colOK ? Q[(r0    ) * D + colB] : 0.f;
        bq0.y = colOK ? Q[(r0 + 1) * D + colB] : 0.f;
        bq1.x = colOK ? Q[(r1    ) * D + colB] : 0.f;
        bq1.y = colOK ? Q[(r1 + 1) * D + colB] : 0.f;
        bq2.x = colOK ? Q[(r2    ) * D + colB] : 0.f;
        float q2y = Q[(((r2 + 1) < D) ? (r2 + 1) : (D - 1)) * D + colB];
        bq2.y = (colOK && ((r2 + 1) < D)) ? q2y : 0.f; // K=11 zero pad
    }
    const float Wl = colOK ? W[colB] : 0.f;
    const float bb = bsc[0];

    // ---- loop-invariant lane constants ----
    const int  c       = 2 * half;                 // A-fragment column base
    const bool aTailOK = (c + 9) < D;              // false only for half==1
    const int  offA    = l15 * D + c;              // a0.x element offset
    const int  offAy   = offA + (aTailOK ? 9 : 8); // a2.y (clamped col for pad)
    const int  offC0   = (8 * half) * D + colB;    // xc[0]; xc[r] at +r*D
    const int  oOff    = (lane & 7) | ((lane & 16) >> 1);
    const bool doStore = (lane & 8) == 0;          // 16 writer lanes
    const int  lastRow = Nrows - 1;

    // precomputed ds_bpermute byte indices + keep/send selectors
    const int  i1 = (lane ^ 1) << 2;
    const int  i2 = (lane ^ 2) << 2;
    const int  i4 = (lane ^ 4) << 2;
    const int  i8 = (lane ^ 8) << 2;
    const bool h1 = (lane & 1) != 0;
    const bool h2 = (lane & 2) != 0;
    const bool h4 = (lane & 4) != 0;

    for (int t = wid; t < numTiles; t += nW) {
        const int base = t << 4;                   // 16 rows/tile; 32-bit safe
        if (t + nW < numTiles)
            __builtin_prefetch(X + (size_t)(t + nW) * 16 * D, 0, 1);

        if (base + 16 <= Nrows) {
            // ---------- fast path: full tile, invariant offsets ----------
            const float* Xb = X + (size_t)base * D;
            v2f a0, a1, a2;
            a0.x = Xb[offA];     a0.y = Xb[offA + 1];
            a1.x = Xb[offA + 4]; a1.y = Xb[offA + 5];
            a2.x = Xb[offA + 8];
            {
                float v = Xb[offAy];
                a2.y = aTailOK ? v : 0.f;          // K=11 zero pad
            }
            float xc[8];
            #pragma unroll
            for (int r = 0; r < 8; ++r) {          // r*D folds into IOFFSET
                float v = Xb[offC0 + r * D];
                xc[r] = colOK ? v : 0.f;
            }

            float f = tile_body(a0, a1, a2, xc, bq0, bq1, bq2, Wl,
                                h1, h2, h4, i1, i2, i4, i8);
            if (doStore)
                out[base + oOff] = f + bb;
        } else {
            // ---------- tail tile: clamped rows, guarded store ----------
            int rowA = base + l15;
            if (rowA > lastRow) rowA = lastRow;
            const float* xr = X + (size_t)rowA * D;
            v2f a0, a1, a2;
            a0.x = xr[c];      a0.y = xr[c + 1];
            a1.x = xr[c + 4];  a1.y = xr[c + 5];
            a2.x = xr[c + 8];
            {
                float v = xr[aTailOK ? (c + 9) : (D - 1)];
                a2.y = aTailOK ? v : 0.f;
            }
            float xc[8];
            #pragma unroll
            for (int r = 0; r < 8; ++r) {
                int row = base + r + 8 * half;
                if (row > lastRow) row = lastRow;
                float v = X[(size_t)row * D + colB];
                xc[r] = colOK ? v : 0.f;
            }

            float f = tile_body(a0, a1, a2, xc, bq0, bq1, bq2, Wl,
                                h1, h2, h4, i1, i2, i4, i8);
            const int orow = base + oOff;
            if (doStore && orow < Nrows)
                out[orow] = f + bb;
        }
    }
}

extern "C" void kernel_launch(void* const* d_in, const int* in_sizes, int n_in,
                              void* d_out, int out_size, void* d_ws, size_t ws_size,
                              hipStream_t stream) {
    (void)n_in; (void)in_sizes; (void)d_ws; (void)ws_size;
    const float* X = (const float*)d_in[0];
    const float* b = (const float*)d_in[1];
    const float* W = (const float*)d_in[2];
    const float* Q = (const float*)d_in[3];
    float* out = (float*)d_out;

    const int Nrows = out_size;                 // output is [1, N]
    const int numTiles = (Nrows + 15) >> 4;

    const int threads = 256;                    // 8 wave32 per block
    const int wavesPerBlock = threads / 32;
    long need = ((long)numTiles + wavesPerBlock - 1) / wavesPerBlock;
    int blocks = (need < 8192) ? (int)need : 8192;
    if (blocks < 1) blocks = 1;

    Quadratic_77936476553490_kernel<<<blocks, threads, 0, stream>>>(
        X, b, W, Q, out, Nrows, numTiles);
}